// PPGCN_25924422598908
// MI455X (gfx1250) — compile-verified
//
#include <hip/hip_runtime.h>
#include <hip/hip_bf16.h>
#include <stdint.h>

#define BLOCK 256
#define TPB   8   // tiles processed per block in the async-staged kernel

#define AS1 __attribute__((address_space(1)))
#define AS3 __attribute__((address_space(3)))

typedef int v4i __attribute__((ext_vector_type(4)));
typedef v4i AS1* gp128_t;   // global (AS1) pointer to 128-bit vector
typedef v4i AS3* lp128_t;   // LDS (AS3) pointer to 128-bit vector

#if defined(__has_builtin)
#if __has_builtin(__builtin_amdgcn_global_load_async_to_lds_b128)
#define HAVE_ASYNC_LDS 1
#endif
#if __has_builtin(__builtin_amdgcn_s_wait_asynccnt)
#define HAVE_WAIT_ASYNC_BUILTIN 1
#endif
#endif

__device__ __forceinline__ void wait_async0() {
#ifdef HAVE_WAIT_ASYNC_BUILTIN
  __builtin_amdgcn_s_wait_asynccnt(0);
#else
  asm volatile("s_wait_asynccnt 0" ::: "memory");
#endif
}

// Per-wave32 segmented reduction of (p, sg): sorted segment ids, so each
// segment occupies a contiguous lane range. Fast path when the whole wave
// holds one segment (common: avg segment length 64 vs 32-lane tiles):
// butterfly sum + single atomic. Otherwise a segmented inclusive scan and
// one atomic per segment-run tail. All 32 lanes execute the shuffles;
// `valid` only gates the atomics (invalid lanes carry sg=-1, p=0).
__device__ __forceinline__ void seg_scan_flush(float p, int sg, int lane,
                                               bool valid, float* agg) {
  int sgFirst = __shfl(sg, 0, 32);
  int sgLast  = __shfl(sg, 31, 32);
  if (sgFirst == sgLast) {            // wave-uniform branch
#pragma unroll
    for (int d = 16; d >= 1; d >>= 1) p += __shfl_xor(p, d, 32);
    if (valid && lane == 31) atomicAdd(&agg[sg], p);
    return;
  }
#pragma unroll
  for (int d = 1; d < 32; d <<= 1) {
    float y  = __shfl_up(p, d, 32);
    int   sy = __shfl_up(sg, d, 32);
    if (lane >= d && sy == sg) p += y;
  }
  int snext = __shfl_down(sg, 1, 32);
  if (valid && ((lane == 31) || (snext != sg)))
    atomicAdd(&agg[sg], p);
}

#ifdef HAVE_ASYNC_LDS
// Async-staged edge accumulate: each block pipelines TPB tiles of 256 edges
// through LDS using global_load_async_to_lds_b128 with double buffering.
// Threads 0..63 stage weights, 64..127 stage neighbor_idx, 128..191 stage
// segment_ids (64 x b128 per 1KB array). Only full tiles reach this kernel.
// Defined first in the file so the disasm snippet shows the async ops.
__global__ __launch_bounds__(BLOCK)
void accum_async(const float* __restrict__ vals,
                 const float* __restrict__ ew,
                 const int* __restrict__ nidx,
                 const int* __restrict__ seg,
                 float* __restrict__ agg,
                 int n_tiles, int iter, const int* __restrict__ nt) {
  if (iter >= *nt) return;

  __shared__ __align__(16) float s_w[2][BLOCK];
  __shared__ __align__(16) int   s_i[2][BLOCK];
  __shared__ __align__(16) int   s_s[2][BLOCK];

  const int tid   = threadIdx.x;
  const int lane  = tid & 31;
  const int grp   = tid >> 6;        // 0..3
  const int j     = (tid & 63) * 4;  // element offset, 4 elems per b128
  const int tile0 = blockIdx.x * TPB;

  int ntiles_here = n_tiles - tile0;
  if (ntiles_here <= 0) return;           // uniform per block
  if (ntiles_here > TPB) ntiles_here = TPB;

  auto issue = [&](int gt, int b) {
    const size_t base = (size_t)gt * BLOCK;
    if (grp == 0) {
      __builtin_amdgcn_global_load_async_to_lds_b128(
          (gp128_t)(ew + base + j), (lp128_t)&s_w[b][j], 0, 0);
    } else if (grp == 1) {
      __builtin_amdgcn_global_load_async_to_lds_b128(
          (gp128_t)(nidx + base + j), (lp128_t)&s_i[b][j], 0, 0);
    } else if (grp == 2) {
      __builtin_amdgcn_global_load_async_to_lds_b128(
          (gp128_t)(seg + base + j), (lp128_t)&s_s[b][j], 0, 0);
    }
  };

  issue(tile0, 0);
  for (int t = 0; t < ntiles_here; ++t) {
    wait_async0();        // this wave's outstanding async load = tile t
    __syncthreads();      // all waves' tile-t staging complete
    if (t + 1 < ntiles_here) issue(tile0 + t + 1, (t + 1) & 1);
    const int b = t & 1;
    float w  = s_w[b][tid];
    int   si = s_i[b][tid];
    int   sg = s_s[b][tid];
    float p  = w * vals[si];            // values table is L2-resident (400KB)
    seg_scan_flush(p, sg, lane, true, agg);
    __syncthreads();      // done reading buffer b before it is overwritten
  }
}
#endif

__global__ void zero_f32(float* __restrict__ p, int n) {
  int i = blockIdx.x * BLOCK + threadIdx.x;
  if (i < n) p[i] = 0.0f;
}

// Direct-load edge accumulate (fallback + tail handling).
__global__ void accum_direct(const float* __restrict__ vals,
                             const float* __restrict__ ew,
                             const int* __restrict__ nidx,
                             const int* __restrict__ seg,
                             float* __restrict__ agg,
                             int start, int n_edges,
                             int iter, const int* __restrict__ nt) {
  if (iter >= *nt) return;
  int e = start + blockIdx.x * BLOCK + threadIdx.x;
  int lane = threadIdx.x & 31;
  bool valid = e < n_edges;
  int   sg = valid ? seg[e] : -1;
  float p  = valid ? ew[e] * vals[nidx[e]] : 0.0f;
  seg_scan_flush(p, sg, lane, valid, agg);
}

// out[i] = sigmoid(agg[i]) while iter < n_times, else identity pass-through.
__global__ void node_update(const float* __restrict__ agg,
                            const float* __restrict__ prev,
                            float* __restrict__ out,
                            int n, int iter, const int* __restrict__ nt) {
  int i = blockIdx.x * BLOCK + threadIdx.x;
  if (i >= n) return;
  if (iter < *nt) {
    float a = agg[i];
    out[i] = 1.0f / (1.0f + __expf(-a));
  } else {
    out[i] = prev[i];
  }
}

extern "C" void kernel_launch(void* const* d_in, const int* in_sizes, int n_in,
                              void* d_out, int out_size, void* d_ws, size_t ws_size,
                              hipStream_t stream) {
  const float* values = (const float*)d_in[0];
  const float* ew     = (const float*)d_in[1];
  const int*   nidx   = (const int*)d_in[2];
  const int*   seg    = (const int*)d_in[3];
  const int*   nt     = (const int*)d_in[4];
  const int n_nodes = in_sizes[0];
  const int n_edges = in_sizes[1];

  float* agg  = (float*)d_ws;          // [n_nodes]
  float* buf0 = agg + n_nodes;         // [n_nodes]
  float* buf1 = buf0 + n_nodes;        // [n_nodes]
  float* out  = (float*)d_out;

  const int nodeBlocks = (n_nodes + BLOCK - 1) / BLOCK;
  const int nFullTiles = n_edges / BLOCK;
  const int tailStart  = nFullTiles * BLOCK;

  const float* src = values;
  for (int it = 0; it < 3; ++it) {
    float* dst = (it == 0) ? buf0 : (it == 1) ? buf1 : out;

    zero_f32<<<nodeBlocks, BLOCK, 0, stream>>>(agg, n_nodes);

#ifdef HAVE_ASYNC_LDS
    if (nFullTiles > 0) {
      int gb = (nFullTiles + TPB - 1) / TPB;
      accum_async<<<gb, BLOCK, 0, stream>>>(src, ew, nidx, seg, agg,
                                            nFullTiles, it, nt);
    }
    if (tailStart < n_edges) {
      int rem = n_edges - tailStart;
      accum_direct<<<(rem + BLOCK - 1) / BLOCK, BLOCK, 0, stream>>>(
          src, ew, nidx, seg, agg, tailStart, n_edges, it, nt);
    }
#else
    accum_direct<<<(n_edges + BLOCK - 1) / BLOCK, BLOCK, 0, stream>>>(
        src, ew, nidx, seg, agg, 0, n_edges, it, nt);
#endif

    node_update<<<nodeBlocks, BLOCK, 0, stream>>>(agg, src, dst, n_nodes, it, nt);
    src = dst;
  }
}